// DeepSeekMOE_85624468013211
// MI455X (gfx1250) — compile-verified
//
#include <hip/hip_runtime.h>
#include <hip/hip_bf16.h>

// ---------------- problem constants (match reference) ----------------
#define NTOK   1024      // B*T = 2*512
#define DIM    512
#define INTER  64
#define NSH    2
#define NROUT  64
#define NEXP   66        // NSH + NROUT
#define TOPK   6
#define KSLOT  8         // TOPK + NSH
#define NREC   (NTOK * KSLOT)   // 8192

typedef float v2f __attribute__((ext_vector_type(2)));
typedef float v8f __attribute__((ext_vector_type(8)));
typedef int   v4i __attribute__((ext_vector_type(4)));

#if __has_builtin(__builtin_amdgcn_global_load_async_to_lds_b128) && \
    __has_builtin(__builtin_amdgcn_s_wait_asynccnt)
#define USE_ASYNC_LDS 1
typedef __attribute__((address_space(1))) v4i gv4i;   // global (device) v4i
typedef __attribute__((address_space(3))) v4i lv4i;   // LDS v4i
#else
#define USE_ASYNC_LDS 0
#endif

// ---------------- kernel 1: zero output + counters ----------------
__global__ void moe_zero_kernel(float* __restrict__ out, int n_out,
                                int* __restrict__ ints, int n_ints) {
    int i = blockIdx.x * blockDim.x + threadIdx.x;
    if (i < n_out)  out[i] = 0.0f;
    if (i < n_ints) ints[i] = 0;
}

// ---------------- kernel 2: gate (softmax + top-6) ----------------
// one block of 64 threads per token; thread e owns routed expert e
__global__ void moe_gate_kernel(const float* __restrict__ x,
                                const float* __restrict__ gw,
                                const float* __restrict__ gbias,
                                float* __restrict__ wgts,  // [NREC] combine weights
                                int*   __restrict__ eids,  // [NREC] expert per (tok,slot)
                                int*   __restrict__ counts) {
    const int t = blockIdx.x;     // token
    const int e = threadIdx.x;    // routed expert 0..63
    __shared__ float xs[DIM];
    __shared__ float sc[NROUT];
    __shared__ float bs[NROUT];

    const float* xr = x + (size_t)t * DIM;
    for (int i = e; i < DIM; i += 64) xs[i] = xr[i];
    __syncthreads();

    const float* gr = gw + (size_t)e * DIM;
    float acc = 0.0f;
    for (int d = 0; d < DIM; d += 4) {
        acc += xs[d] * gr[d] + xs[d + 1] * gr[d + 1]
             + xs[d + 2] * gr[d + 2] + xs[d + 3] * gr[d + 3];
    }
    sc[e] = acc;
    __syncthreads();

    if (e == 0) {
        // softmax over 64 (fp32, matches reference)
        float m = sc[0];
        for (int i = 1; i < NROUT; i++) m = fmaxf(m, sc[i]);
        float s = 0.0f;
        for (int i = 0; i < NROUT; i++) { sc[i] = __expf(sc[i] - m); s += sc[i]; }
        const float inv = 1.0f / s;
        for (int i = 0; i < NROUT; i++) { sc[i] *= inv; bs[i] = sc[i] + gbias[i]; }
        // shared experts: slots 0..1, weight 1.0
        for (int s2 = 0; s2 < NSH; s2++) {
            wgts[t * KSLOT + s2] = 1.0f;
            eids[t * KSLOT + s2] = s2;
        }
        // top-6 on biased scores (strict > keeps lowest index on ties, like top_k)
        for (int k = 0; k < TOPK; k++) {
            int bi = 0; float bv = bs[0];
            for (int i = 1; i < NROUT; i++) { if (bs[i] > bv) { bv = bs[i]; bi = i; } }
            wgts[t * KSLOT + NSH + k] = sc[bi] * 1.0f;  // ROUTE_SCALE = 1
            eids[t * KSLOT + NSH + k] = NSH + bi;
            bs[bi] = -1e30f;
        }
    }
    __syncthreads();
    if (e < KSLOT) atomicAdd(&counts[eids[t * KSLOT + e]], 1);
}

// ---------------- kernel 3: exclusive scan over 66 experts ----------------
__global__ void moe_scan_kernel(const int* __restrict__ counts,
                                int* __restrict__ offsets) {
    if (threadIdx.x == 0 && blockIdx.x == 0) {
        int acc = 0;
        for (int i = 0; i < NEXP; i++) { offsets[i] = acc; acc += counts[i]; }
        offsets[NEXP] = acc;
    }
}

// ---------------- kernel 4: scatter (token,slot) records into expert bins ----
__global__ void moe_scatter_kernel(const int* __restrict__ eids,
                                   const int* __restrict__ offsets,
                                   int* __restrict__ cursors,
                                   int* __restrict__ recs) {
    int ts = blockIdx.x * blockDim.x + threadIdx.x;  // 0..NREC-1
    if (ts >= NREC) return;
    int e = eids[ts];
    int pos = atomicAdd(&cursors[e], 1);
    recs[offsets[e] + pos] = ts;
}

// ---------------- kernel 5: expert FFN via f32 WMMA ----------------
// grid: (NEXP, 64 tiles); block: 256 threads = 8 wave32s.
// Per block: 16 gathered tokens.
//   GEMM1: H1/H3[16x64] = X[16x512] @ W13[512x64]   (waves 0-3: w1, 4-7: w3)
//          two independent WMMA accumulator chains (ILP across the XDL hazard)
//   act  : Hs = silu(H1) * H3
//   GEMM2: O[16x512] = Hs[16x64] @ W2[64x512], two N-tiles in flight per wave,
//          scaled + atomic combine.
#define XPAD 4   // LDS padding: row stride 516 dwords -> conflict-free column reads
#define HPAD 4
__global__ __launch_bounds__(256) void moe_ffn_kernel(
        const float* __restrict__ x,
        const float* __restrict__ w1,
        const float* __restrict__ w2,
        const float* __restrict__ w3,
        const int*   __restrict__ recs,
        const int*   __restrict__ offsets,
        const int*   __restrict__ counts,
        const float* __restrict__ wgts,
        float* __restrict__ out) {
    const int e    = blockIdx.x;
    const int tile = blockIdx.y;
    const int count = counts[e];
    if (tile * 16 >= count) return;      // uniform across the block

    __shared__ float Xs[16][DIM + XPAD];
    __shared__ float H13[2][16][INTER + HPAD];
    __shared__ float Hs[16][INTER + HPAD];
    __shared__ int   tok[16];
    __shared__ float rw[16];
    __shared__ int   valid[16];

    const int tid = threadIdx.x;
    const int base = offsets[e] + tile * 16;

    if (tid < 16) {
        const int r  = tid;
        const int ok = (tile * 16 + r) < count;
        const int rec = ok ? recs[base + r] : 0;
        tok[r]   = rec >> 3;        // rec = token*KSLOT + slot
        rw[r]    = ok ? wgts[rec] : 0.0f;
        valid[r] = ok;
    }
    __syncthreads();

    // ---- stage X rows into LDS (async LDS DMA if available; zeros otherwise) ----
    {
        const int row = tid >> 4;   // 0..15
        const int l   = tid & 15;   // 0..15 lanes per row
#if USE_ASYNC_LDS
        if (valid[row]) {
            const float* src = x + (size_t)tok[row] * DIM;
            for (int j = 0; j < 8; j++) {
                const float* gp = src + 4 * (l + 16 * j);
                float*       lp = &Xs[row][4 * (l + 16 * j)];
                // low 32 bits of a generic LDS pointer == LDS byte address
                __builtin_amdgcn_global_load_async_to_lds_b128(
                    (gv4i*)(uintptr_t)gp,
                    (lv4i*)(unsigned)(uintptr_t)lp,
                    0, 0);
            }
        } else {
            for (int j = 0; j < 8; j++) {
                Xs[row][4 * (l + 16 * j) + 0] = 0.0f;
                Xs[row][4 * (l + 16 * j) + 1] = 0.0f;
                Xs[row][4 * (l + 16 * j) + 2] = 0.0f;
                Xs[row][4 * (l + 16 * j) + 3] = 0.0f;
            }
        }
        __builtin_amdgcn_s_wait_asynccnt(0);
#else
        if (valid[row]) {
            const float4* src = (const float4*)(x + (size_t)tok[row] * DIM);
            for (int j = 0; j < 8; j++) {
                float4 v = src[l + 16 * j];
                Xs[row][4 * (l + 16 * j) + 0] = v.x;
                Xs[row][4 * (l + 16 * j) + 1] = v.y;
                Xs[row][4 * (l + 16 * j) + 2] = v.z;
                Xs[row][4 * (l + 16 * j) + 3] = v.w;
            }
        } else {
            for (int j = 0; j < 8; j++) {
                Xs[row][4 * (l + 16 * j) + 0] = 0.0f;
                Xs[row][4 * (l + 16 * j) + 1] = 0.0f;
                Xs[row][4 * (l + 16 * j) + 2] = 0.0f;
                Xs[row][4 * (l + 16 * j) + 3] = 0.0f;
            }
        }
#endif
    }
    __syncthreads();

    const int wave = tid >> 5;      // 0..7
    const int lane = tid & 31;
    const int lm   = lane & 15;     // column / row-in-half
    const int hi   = lane >> 4;     // 0: lanes 0-15 (K 0,1), 1: lanes 16-31 (K 2,3)

    // ---- GEMM1: waves 0-3 -> h1 (w1), waves 4-7 -> h3 (w3); n-tile = wave&3 ----
    {
        const int  hsel = wave >> 2;
        const int  n    = wave & 3;
        const float* W  = (hsel == 0 ? w1 : w3) + (size_t)e * DIM * INTER;

        v8f acc0 = {}, acc1 = {};      // two independent XDL chains
        for (int k = 0; k < DIM; k += 8) {
            if ((k & 127) == 0)
                __builtin_prefetch(&W[(size_t)(k + 128) * INTER + n * 16 + lm], 0, 1);
            v2f a0, b0, a1, b1;
            // A: 16x4 f32 fragments (row lm; K = k+2*hi, k+2*hi+1)
            a0.x = Xs[lm][k     + 2 * hi];
            a0.y = Xs[lm][k     + 2 * hi + 1];
            a1.x = Xs[lm][k + 4 + 2 * hi];
            a1.y = Xs[lm][k + 4 + 2 * hi + 1];
            // B: 4x16 f32 fragments (col n*16+lm; K rows mirror A)
            b0.x = W[(size_t)(k     + 2 * hi)     * INTER + n * 16 + lm];
            b0.y = W[(size_t)(k     + 2 * hi + 1) * INTER + n * 16 + lm];
            b1.x = W[(size_t)(k + 4 + 2 * hi)     * INTER + n * 16 + lm];
            b1.y = W[(size_t)(k + 4 + 2 * hi + 1) * INTER + n * 16 + lm];
            acc0 = __builtin_amdgcn_wmma_f32_16x16x4_f32(
                       false, a0, false, b0, (short)0, acc0, false, false);
            acc1 = __builtin_amdgcn_wmma_f32_16x16x4_f32(
                       false, a1, false, b1, (short)0, acc1, false, false);
        }
        // C/D layout: VGPR r -> row r (lanes 0-15) / row 8+r (lanes 16-31)
        for (int r = 0; r < 8; r++) {
            const int row = hi ? (8 + r) : r;
            H13[hsel][row][n * 16 + lm] = acc0[r] + acc1[r];
        }
    }
    __syncthreads();

    // ---- activation: Hs = silu(h1) * h3 (hw v_rcp_f32, ~1 ulp) ----
    for (int i = tid; i < 16 * INTER; i += 256) {
        const int row = i >> 6, c = i & 63;
        const float h1 = H13[0][row][c];
        const float h3 = H13[1][row][c];
        const float sig = __builtin_amdgcn_rcpf(1.0f + __expf(-h1));
        Hs[row][c] = h1 * sig * h3;
    }
    __syncthreads();

    // ---- GEMM2: O[16x512]; each wave runs two N-tiles concurrently ----
    const float* W2 = w2 + (size_t)e * INTER * DIM;
    for (int jp = 0; jp < 2; jp++) {
        const int n2a = wave + 8 * (2 * jp);
        const int n2b = wave + 8 * (2 * jp + 1);
        __builtin_prefetch(&W2[(size_t)(2 * hi) * DIM + n2a * 16 + lm], 0, 1);
        v8f accA = {}, accB = {};
        for (int k = 0; k < INTER; k += 4) {
            v2f a, ba, bb;
            a.x = Hs[lm][k + 2 * hi];
            a.y = Hs[lm][k + 2 * hi + 1];
            ba.x = W2[(size_t)(k + 2 * hi)     * DIM + n2a * 16 + lm];
            ba.y = W2[(size_t)(k + 2 * hi + 1) * DIM + n2a * 16 + lm];
            bb.x = W2[(size_t)(k + 2 * hi)     * DIM + n2b * 16 + lm];
            bb.y = W2[(size_t)(k + 2 * hi + 1) * DIM + n2b * 16 + lm];
            accA = __builtin_amdgcn_wmma_f32_16x16x4_f32(
                       false, a, false, ba, (short)0, accA, false, false);
            accB = __builtin_amdgcn_wmma_f32_16x16x4_f32(
                       false, a, false, bb, (short)0, accB, false, false);
        }
        for (int r = 0; r < 8; r++) {
            const int row = hi ? (8 + r) : r;
            if (valid[row]) {
                const float w = rw[row];
                atomicAdd(&out[(size_t)tok[row] * DIM + n2a * 16 + lm], accA[r] * w);
                atomicAdd(&out[(size_t)tok[row] * DIM + n2b * 16 + lm], accB[r] * w);
            }
        }
    }
}

// ---------------- launcher ----------------
extern "C" void kernel_launch(void* const* d_in, const int* in_sizes, int n_in,
                              void* d_out, int out_size, void* d_ws, size_t ws_size,
                              hipStream_t stream) {
    (void)in_sizes; (void)n_in; (void)out_size; (void)ws_size;
    const float* x  = (const float*)d_in[0];
    const float* gw = (const float*)d_in[1];
    const float* gb = (const float*)d_in[2];
    const float* w1 = (const float*)d_in[3];
    const float* w2 = (const float*)d_in[4];
    const float* w3 = (const float*)d_in[5];
    float* out = (float*)d_out;

    // workspace layout (all 4-byte aligned)
    int* counts  = (int*)d_ws;              // 66
    int* cursors = counts + NEXP;           // 66
    int* offsets = cursors + NEXP;          // 67
    int* eids    = offsets + (NEXP + 2);    // 8192 (pad offsets to 68)
    int* recs    = eids + NREC;             // 8192
    float* wgts  = (float*)(recs + NREC);   // 8192

    const int n_out = NTOK * DIM;           // 524288
    moe_zero_kernel<<<(n_out + 255) / 256, 256, 0, stream>>>(
        out, n_out, counts, 2 * NEXP);      // zero counts+cursors together

    moe_gate_kernel<<<NTOK, 64, 0, stream>>>(x, gw, gb, wgts, eids, counts);

    moe_scan_kernel<<<1, 1, 0, stream>>>(counts, offsets);

    moe_scatter_kernel<<<NREC / 256, 256, 0, stream>>>(eids, offsets, cursors, recs);

    dim3 grid(NEXP, NTOK / 16);             // 66 x 64; empty tiles exit early
    moe_ffn_kernel<<<grid, 256, 0, stream>>>(
        x, w1, w2, w3, recs, offsets, counts, wgts, out);
}